// MBDCompressor3D_5128190952090
// MI455X (gfx1250) — compile-verified
//
#include <hip/hip_runtime.h>
#include <cstddef>

typedef float v2f __attribute__((ext_vector_type(2)));
typedef float v8f __attribute__((ext_vector_type(8)));

#define QPTS 262144
#define EPSF 1e-8f
#define PB 128          // points per block
#define NTHREADS 256    // 8 wave32

// ---- LDS layout (float offsets), total 40256 floats = ~157 KB dynamic LDS ----
#define OFF_GPHI   0        // [128][64] unnormalized phi weights
#define OFF_GPSI   8192     // [128][64] unnormalized psi weights
#define OFF_SPHI   16384    // [128] 1/(rowsum+eps) for phi
#define OFF_SPSI   16512    // [128] 1/(rowsum+eps) for psi
#define OFF_PREC   16640    // [128][12]: mu0..2, p00,p01,p02,p11,p12,p22
#define OFF_B      18176    // [4 nt][16 kk][32 lane][2]  B pre-swizzled into WMMA fragments
#define OFF_C      22272    // [16 kk][32 lane][2]        C fragments
#define OFF_W1     23296    // [2 nt][8 kk][32 lane][2]   W1 fragments
#define OFF_W2     24320    // [2 nt][8 kk][32 lane][2]   W2 fragments
#define OFF_BIAS   25344    // [64]: b1[32], b2 padded[32]
#define OFF_COORD  25408    // [128][4]
#define OFF_MB     25920    // [128][64] moving_basis tile (padded)
#define OFF_MC     34112    // [128][16] moving_coeff tile (padded)
#define OFF_IO     36160    // [128][32] mlp_in, later overwritten with h
#define SMEM_FLOATS 40256

extern __shared__ float smem[];

__global__ __launch_bounds__(NTHREADS)
void mbd_compressor_kernel(const float* __restrict__ coords,
                           const float* __restrict__ coeff_mu,
                           const float* __restrict__ coeff_log_s,
                           const float* __restrict__ coeff_q,
                           const float* __restrict__ basis_mu,
                           const float* __restrict__ basis_log_s,
                           const float* __restrict__ basis_q,
                           const float* __restrict__ Cg,
                           const float* __restrict__ Bg,
                           const float* __restrict__ W1g,
                           const float* __restrict__ b1g,
                           const float* __restrict__ W2g,
                           const float* __restrict__ b2g,
                           float* __restrict__ out)
{
  const int tid = threadIdx.x;
  const size_t qblk = (size_t)blockIdx.x * PB;

  // ================= Phase 0: stage weights / precision / coords =================
  if (tid < 128) {
    const int  k        = tid & 63;
    const bool is_basis = tid >= 64;
    const float* mu = is_basis ? basis_mu    : coeff_mu;
    const float* ls = is_basis ? basis_log_s : coeff_log_s;
    const float* qp = is_basis ? basis_q     : coeff_q;

    const float sx = __expf(ls[k*3+0]);
    const float sy = __expf(ls[k*3+1]);
    const float sz = __expf(ls[k*3+2]);
    const float ix = 1.f/(sx*sx + EPSF);
    const float iy = 1.f/(sy*sy + EPSF);
    const float iz = 1.f/(sz*sz + EPSF);

    float qw = qp[k*4+0], qx = qp[k*4+1], qy = qp[k*4+2], qz = qp[k*4+3];
    const float qn = 1.f/(sqrtf(qw*qw + qx*qx + qy*qy + qz*qz) + EPSF);
    qw *= qn; qx *= qn; qy *= qn; qz *= qn;

    const float r00 = 1.f - 2.f*(qy*qy + qz*qz), r01 = 2.f*(qx*qy - qw*qz), r02 = 2.f*(qx*qz + qw*qy);
    const float r10 = 2.f*(qx*qy + qw*qz), r11 = 1.f - 2.f*(qx*qx + qz*qz), r12 = 2.f*(qy*qz - qw*qx);
    const float r20 = 2.f*(qx*qz - qw*qy), r21 = 2.f*(qy*qz + qw*qx), r22 = 1.f - 2.f*(qx*qx + qy*qy);

    // prec = R diag(inv) R^T (symmetric)
    float* pc = smem + OFF_PREC + tid*12;
    pc[0] = mu[k*3+0]; pc[1] = mu[k*3+1]; pc[2] = mu[k*3+2];
    pc[3] = r00*r00*ix + r01*r01*iy + r02*r02*iz;   // p00
    pc[4] = r00*r10*ix + r01*r11*iy + r02*r12*iz;   // p01
    pc[5] = r00*r20*ix + r01*r21*iy + r02*r22*iz;   // p02
    pc[6] = r10*r10*ix + r11*r11*iy + r12*r12*iz;   // p11
    pc[7] = r10*r20*ix + r11*r21*iy + r12*r22*iz;   // p12
    pc[8] = r20*r20*ix + r21*r21*iy + r22*r22*iz;   // p22
  }

  // Stage B/C/W1/W2 pre-swizzled into WMMA B-fragment order:
  // fragment element (nt, kk, lane, e) <- weight[k = 4*kk + 2*(lane>>4) + e][col = 16*nt + (lane&15)]
  for (int i = tid; i < 4096; i += NTHREADS) {          // B: 4 ntiles x 16 ksteps
    const int e = i & 1, p = i >> 1;
    const int ln = p & 31, kk = (p >> 5) & 15, nt = p >> 9;
    const int k = 4*kk + 2*(ln >> 4) + e;
    const int c = 16*nt + (ln & 15);
    smem[OFF_B + i] = (c < 54) ? Bg[k*54 + c] : 0.f;
  }
  for (int i = tid; i < 1024; i += NTHREADS) {          // C: 1 ntile x 16 ksteps
    const int e = i & 1, p = i >> 1;
    const int ln = p & 31, kk = p >> 5;
    const int k = 4*kk + 2*(ln >> 4) + e;
    const int c = ln & 15;
    smem[OFF_C + i] = (c < 2) ? Cg[k*2 + c] : 0.f;
  }
  for (int i = tid; i < 1024; i += NTHREADS) {          // W1: 2 ntiles x 8 ksteps
    const int e = i & 1, p = i >> 1;
    const int ln = p & 31, kk = (p >> 5) & 7, nt = p >> 8;
    const int k = 4*kk + 2*(ln >> 4) + e;
    const int c = 16*nt + (ln & 15);
    smem[OFF_W1 + i] = (k < 30) ? W1g[k*32 + c] : 0.f;
  }
  for (int i = tid; i < 1024; i += NTHREADS) {          // W2: 2 ntiles x 8 ksteps
    const int e = i & 1, p = i >> 1;
    const int ln = p & 31, kk = (p >> 5) & 7, nt = p >> 8;
    const int k = 4*kk + 2*(ln >> 4) + e;
    const int c = 16*nt + (ln & 15);
    smem[OFF_W2 + i] = (c < 27) ? W2g[k*32 + c] : 0.f;
  }
  if (tid < 64) smem[OFF_BIAS+tid] = (tid < 32) ? b1g[tid] : ((tid-32) < 27 ? b2g[tid-32] : 0.f);
  if (tid < PB) {
    const size_t p = qblk + tid;
    smem[OFF_COORD + tid*4 + 0] = coords[p*3+0];
    smem[OFF_COORD + tid*4 + 1] = coords[p*3+1];
    smem[OFF_COORD + tid*4 + 2] = coords[p*3+2];
    smem[OFF_COORD + tid*4 + 3] = 0.f;
  }
  __syncthreads();

  // ========== Phase 1: unnormalized gaussian weights + inverse row sums ==========
  // Row-normalization commutes with the GEMMs, so fold it into a post-GEMM scale.
  {
    const int half = tid >> 7;      // 0 = phi (coeff), 1 = psi (basis)
    const int r    = tid & 127;
    const float px = smem[OFF_COORD + r*4 + 0];
    const float py = smem[OFF_COORD + r*4 + 1];
    const float pz = smem[OFF_COORD + r*4 + 2];
    float* G = smem + (half ? OFF_GPSI : OFF_GPHI) + r*64;
    const float* pc0 = smem + OFF_PREC + half*64*12;
    float sum = 0.f;
    #pragma unroll 8
    for (int k = 0; k < 64; ++k) {
      const float* P = pc0 + k*12;
      const float d0 = px - P[0], d1 = py - P[1], d2 = pz - P[2];
      const float mah = P[3]*d0*d0 + P[6]*d1*d1 + P[8]*d2*d2
                      + 2.f*(P[4]*d0*d1 + P[5]*d0*d2 + P[7]*d1*d2);
      const float g = __expf(-0.5f*mah);
      G[k] = g;
      sum += g;
    }
    smem[(half ? OFF_SPSI : OFF_SPHI) + r] = 1.f/(sum + EPSF);
  }
  __syncthreads();

  // ================= Phase 2: per-wave WMMA GEMMs (16-row tiles) =================
  const int wv   = tid >> 5;
  const int lane = tid & 31;
  const int r0   = wv * 16;
  const int arow = lane & 15;   // A fragment: row
  const int ah   = lane >> 4;   // A fragment: K sub-offset 2*ah
  const int ch   = lane >> 4;   // C/D fragment: rows v + 8*ch
  const size_t qrow0 = qblk + r0;

  v2f a64[16];

  // ---- moving_basis = rowscale(psi_u) @ B : [16x64]@[64x54->64] ----
  {
    const float* Ap = smem + OFF_GPSI + (r0 + arow)*64 + 2*ah;
    #pragma unroll
    for (int kk = 0; kk < 16; ++kk) a64[kk] = *(const v2f*)&Ap[4*kk];
  }
  float* out_mb = out + (size_t)QPTS*29;
  #pragma unroll
  for (int nt = 0; nt < 4; ++nt) {
    v8f acc = {};
    #pragma unroll
    for (int kk = 0; kk < 16; ++kk) {
      const v2f b = *(const v2f*)&smem[OFF_B + ((nt*16 + kk)*32 + lane)*2];
      acc = __builtin_amdgcn_wmma_f32_16x16x4_f32(false, a64[kk], false, b,
                                                  (short)0, acc, false, false);
    }
    const int col = 16*nt + (lane & 15);
    #pragma unroll
    for (int v = 0; v < 8; ++v) {
      const int rr = v + 8*ch;
      const float val = acc[v] * smem[OFF_SPSI + r0 + rr];
      smem[OFF_MB + (r0+rr)*64 + col] = val;
      if (col < 54) out_mb[(qrow0 + rr)*54 + col] = val;
    }
  }

  // ---- moving_coeff = rowscale(phi_u) @ C : [16x64]@[64x2->16] ----
  {
    const float* Ap = smem + OFF_GPHI + (r0 + arow)*64 + 2*ah;
    #pragma unroll
    for (int kk = 0; kk < 16; ++kk) a64[kk] = *(const v2f*)&Ap[4*kk];
    v8f acc = {};
    #pragma unroll
    for (int kk = 0; kk < 16; ++kk) {
      const v2f b = *(const v2f*)&smem[OFF_C + (kk*32 + lane)*2];
      acc = __builtin_amdgcn_wmma_f32_16x16x4_f32(false, a64[kk], false, b,
                                                  (short)0, acc, false, false);
    }
    const int col = lane & 15;
    float* out_mc = out + (size_t)QPTS*27;
    #pragma unroll
    for (int v = 0; v < 8; ++v) {
      const int rr = v + 8*ch;
      const float val = acc[v] * smem[OFF_SPHI + r0 + rr];
      smem[OFF_MC + (r0+rr)*16 + col] = val;
      if (col < 2) out_mc[(qrow0 + rr)*2 + col] = val;
    }
  }
  __syncthreads();

  // ---- mbd + mlp_in assembly (block-parallel elementwise) ----
  {
    float* out_mbd = out + (size_t)QPTS*83;
    for (int e = tid; e < PB*32; e += NTHREADS) {
      const int r = e >> 5, c = e & 31;
      float v;
      if (c < 27) {
        const float mc0 = smem[OFF_MC + r*16 + 0];
        const float mc1 = smem[OFF_MC + r*16 + 1];
        v = mc0 * smem[OFF_MB + r*64 + c] + mc1 * smem[OFF_MB + r*64 + 27 + c];
        out_mbd[(qblk + r)*27 + c] = v;
      } else if (c < 30) {
        v = smem[OFF_COORD + r*4 + (c - 27)];
      } else {
        v = 0.f;
      }
      smem[OFF_IO + e] = v;   // mlp_in [128][32]
    }
  }
  __syncthreads();

  // ---- MLP layer 1: relu(mlp_in[16x32] @ W1[32x32] + b1) ----
  v2f a32[8];
  {
    const float* Ap = smem + OFF_IO + (r0 + arow)*32 + 2*ah;
    #pragma unroll
    for (int kk = 0; kk < 8; ++kk) a32[kk] = *(const v2f*)&Ap[4*kk];
  }
  v8f hacc[2];
  #pragma unroll
  for (int nt = 0; nt < 2; ++nt) {
    const int col = 16*nt + (lane & 15);
    const float bb = smem[OFF_BIAS + col];
    v8f acc = {bb, bb, bb, bb, bb, bb, bb, bb};
    #pragma unroll
    for (int kk = 0; kk < 8; ++kk) {
      const v2f b = *(const v2f*)&smem[OFF_W1 + ((nt*8 + kk)*32 + lane)*2];
      acc = __builtin_amdgcn_wmma_f32_16x16x4_f32(false, a32[kk], false, b,
                                                  (short)0, acc, false, false);
    }
    hacc[nt] = acc;
  }
  __syncthreads();  // all reads of mlp_in done before overwriting region with h
  #pragma unroll
  for (int nt = 0; nt < 2; ++nt) {
    const int col = 16*nt + (lane & 15);
    #pragma unroll
    for (int v = 0; v < 8; ++v) {
      const int rr = v + 8*ch;
      smem[OFF_IO + (r0+rr)*32 + col] = fmaxf(hacc[nt][v], 0.f);
    }
  }
  __syncthreads();

  // ---- MLP layer 2: h[16x32] @ W2[32x27->32] + b2 -> reconstruction ----
  {
    const float* Ap = smem + OFF_IO + (r0 + arow)*32 + 2*ah;
    #pragma unroll
    for (int kk = 0; kk < 8; ++kk) a32[kk] = *(const v2f*)&Ap[4*kk];
    #pragma unroll
    for (int nt = 0; nt < 2; ++nt) {
      const int col = 16*nt + (lane & 15);
      const float bb = smem[OFF_BIAS + 32 + col];
      v8f acc = {bb, bb, bb, bb, bb, bb, bb, bb};
      #pragma unroll
      for (int kk = 0; kk < 8; ++kk) {
        const v2f b = *(const v2f*)&smem[OFF_W2 + ((nt*8 + kk)*32 + lane)*2];
        acc = __builtin_amdgcn_wmma_f32_16x16x4_f32(false, a32[kk], false, b,
                                                    (short)0, acc, false, false);
      }
      if (col < 27) {
        #pragma unroll
        for (int v = 0; v < 8; ++v) {
          const int rr = v + 8*ch;
          out[(qrow0 + rr)*27 + col] = acc[v];
        }
      }
    }
  }
}

extern "C" void kernel_launch(void* const* d_in, const int* in_sizes, int n_in,
                              void* d_out, int out_size, void* d_ws, size_t ws_size,
                              hipStream_t stream) {
  (void)in_sizes; (void)n_in; (void)out_size; (void)d_ws; (void)ws_size;
  const float* coords      = (const float*)d_in[0];
  const float* coeff_mu    = (const float*)d_in[1];
  const float* coeff_log_s = (const float*)d_in[2];
  const float* coeff_q     = (const float*)d_in[3];
  const float* basis_mu    = (const float*)d_in[4];
  const float* basis_log_s = (const float*)d_in[5];
  const float* basis_q     = (const float*)d_in[6];
  const float* Cg          = (const float*)d_in[7];
  const float* Bg          = (const float*)d_in[8];
  const float* W1g         = (const float*)d_in[9];
  const float* b1g         = (const float*)d_in[10];
  const float* W2g         = (const float*)d_in[11];
  const float* b2g         = (const float*)d_in[12];
  float* out = (float*)d_out;

  dim3 grid(QPTS / PB);
  dim3 block(NTHREADS);
  const size_t smem_bytes = (size_t)SMEM_FLOATS * sizeof(float); // ~157 KB of the 320 KB WGP LDS
  mbd_compressor_kernel<<<grid, block, smem_bytes, stream>>>(
      coords, coeff_mu, coeff_log_s, coeff_q,
      basis_mu, basis_log_s, basis_q,
      Cg, Bg, W1g, b1g, W2g, b2g, out);
}